// RegionProposalNetwork1d_40381282517186
// MI455X (gfx1250) — compile-verified
//
#include <hip/hip_runtime.h>
#include <hip/hip_bf16.h>
#include <cstdint>

// ---------------------------------------------------------------------------
// Model constants
// ---------------------------------------------------------------------------
#define SEQ_L   262144
#define NANCH   5
#define PRE_N   6000
#define POST_N  300
#define NMS_W   192          // ceil(6000/32)=188, rounded up to 192
#define SORT_N  8192

typedef float v2f __attribute__((ext_vector_type(2)));
typedef float v8f __attribute__((ext_vector_type(8)));

// ---------------------------------------------------------------------------
// Depthwise conv1d k=3 (pad==dil, length preserving), dual-source (U-Net concat)
// ---------------------------------------------------------------------------
__global__ void dw3_conv(const float* __restrict__ x0, int c0,
                         const float* __restrict__ x1, int c1,
                         const float* __restrict__ w, const float* __restrict__ b,
                         float* __restrict__ out, int dil, int L, int doRelu)
{
    int C = c0 + c1;
    long i = (long)blockIdx.x * blockDim.x + threadIdx.x;
    if (i >= (long)C * L) return;
    int c = (int)(i / L);
    int l = (int)(i % L);
    const float* row = (c < c0) ? (x0 + (size_t)c * L) : (x1 + (size_t)(c - c0) * L);
    float acc = b[c];
#pragma unroll
    for (int j = 0; j < 3; ++j) {
        int ll = l + (j - 1) * dil;
        if (ll >= 0 && ll < L) acc += w[c * 3 + j] * row[ll];
    }
    if (doRelu) acc = fmaxf(acc, 0.f);
    out[(size_t)c * L + l] = acc;
}

// ---------------------------------------------------------------------------
// Pointwise conv as GEMM with V_WMMA_F32_16X16X4_F32, templated on (CIN, COUT)
// so the K-loop fully unrolls into a straight chain of WMMAs and all range
// guards constant-fold for aligned layers.
//   Y[co,l] = sum_ci W[co,ci] * T[ci,l] + bias[co]
// One wave -> 16(cout) x 16(L) tile.  Weights async-staged to LDS via
// global_load_async_to_lds_b32 (ASYNCcnt path), completed with s_wait_asynccnt.
// ---------------------------------------------------------------------------
template<int CIN, int COUT>
__global__ void __launch_bounds__(128)
pw_conv_wmma(const float* __restrict__ T, const float* __restrict__ W,
             const float* __restrict__ bias, float* __restrict__ Y, int L)
{
    __shared__ float sW[CIN * COUT];
    for (int e = threadIdx.x; e < CIN * COUT; e += 128) {
        uint32_t loff = (uint32_t)(uintptr_t)(&sW[e]);   // low 32 bits = LDS offset
        const float* g = W + e;
        asm volatile("global_load_async_to_lds_b32 %0, %1, off"
                     :: "v"(loff), "v"(g) : "memory");
    }
    asm volatile("s_wait_asynccnt 0" ::: "memory");
    __syncthreads();

    const int wave   = threadIdx.x >> 5;
    const int lane   = threadIdx.x & 31;
    const int hlf    = lane >> 4;          // 0: lanes 0-15, 1: lanes 16-31
    const int lane16 = lane & 15;
    const int n0 = (blockIdx.x * 4 + wave) * 16;   // L tile
    const int m0 = blockIdx.y * 16;                // cout tile
    const int m  = m0 + lane16;
    const int n  = n0 + lane16;
    const bool mok = (COUT % 16 == 0) ? true : (m < COUT);

    v8f acc = {};
    constexpr int kTiles = (CIN + 3) / 4;
#pragma unroll
    for (int kt = 0; kt < kTiles; ++kt) {
        const int kb = kt * 4 + 2 * hlf;
        const bool k0 = (CIN % 4 == 0) ? true : (kb + 0 < CIN);
        const bool k1 = (CIN % 4 == 0) ? true : (kb + 1 < CIN);
        // A fragment: 16x4 (MxK).  VGPR0: K = 2*hlf, VGPR1: K = 2*hlf+1, M = lane16.
        v2f a, bfr;
        a.x = (mok && k0) ? sW[m * CIN + kb + 0] : 0.f;
        a.y = (mok && k1) ? sW[m * CIN + kb + 1] : 0.f;
        // B fragment: 4x16 (KxN).  VGPR r holds K = r + 2*hlf, N = lane16.
        bfr.x = k0 ? T[(size_t)(kb + 0) * L + n] : 0.f;
        bfr.y = k1 ? T[(size_t)(kb + 1) * L + n] : 0.f;
        if (kt + 1 < kTiles)   // stream next K rows through L2 (global_prefetch)
            __builtin_prefetch(&T[(size_t)(kt * 4 + 4) * L + n], 0, 1);
        acc = __builtin_amdgcn_wmma_f32_16x16x4_f32(false, a, false, bfr,
                                                    (short)0, acc, false, false);
    }
#pragma unroll
    for (int r = 0; r < 8; ++r) {          // C/D: VGPR r holds M = r + 8*hlf
        const int mo = m0 + r + 8 * hlf;
        if ((COUT % 16 == 0) || (mo < COUT))
            Y[(size_t)mo * L + n0 + lane16] = acc[r] + bias[mo];
    }
}

// ---------------------------------------------------------------------------
// Per-channel mean over L (SE pooling)
// ---------------------------------------------------------------------------
__global__ void chan_mean(const float* __restrict__ X, float* __restrict__ mean, int L)
{
    __shared__ float red[256];
    const float* row = X + (size_t)blockIdx.x * L;
    float s = 0.f;
    for (int l = threadIdx.x; l < L; l += 256) s += row[l];
    red[threadIdx.x] = s; __syncthreads();
    for (int k = 128; k > 0; k >>= 1) {
        if (threadIdx.x < k) red[threadIdx.x] += red[threadIdx.x + k];
        __syncthreads();
    }
    if (threadIdx.x == 0) mean[blockIdx.x] = red[0] / (float)L;
}

// Tiny SE MLP: s = sigmoid(W2 relu(W1 mean + b1) + b2)
__global__ void se_mlp(const float* __restrict__ mean,
                       const float* __restrict__ w1, const float* __restrict__ b1,
                       const float* __restrict__ w2, const float* __restrict__ b2,
                       float* __restrict__ s, int C, int H)
{
    __shared__ float th[8];
    int t = threadIdx.x;
    if (t < H) {
        float a = b1[t];
        for (int c = 0; c < C; ++c) a += w1[t * C + c] * mean[c];
        th[t] = fmaxf(a, 0.f);
    }
    __syncthreads();
    if (t < C) {
        float a = b2[t];
        for (int h = 0; h < H; ++h) a += w2[t * H + h] * th[h];
        s[t] = 1.f / (1.f + expf(-a));
    }
}

// Fused channel-scale + depthwise spatial attention (ka=7) + relu.
// a = sigmoid(dwconv7(Y*s)); O = relu(Y*s*a).  Neighbor scale == s[c] -> fully fused.
__global__ void attn_apply(const float* __restrict__ Y, const float* __restrict__ s,
                           const float* __restrict__ aw, const float* __restrict__ ab,
                           float* __restrict__ O, int C, int L)
{
    long i = (long)blockIdx.x * blockDim.x + threadIdx.x;
    if (i >= (long)C * L) return;
    int c = (int)(i / L);
    int l = (int)(i % L);
    float sc = s[c];
    float acc = ab[c];
#pragma unroll
    for (int j = 0; j < 7; ++j) {
        int ll = l + j - 3;
        if (ll >= 0 && ll < L) acc += aw[c * 7 + j] * Y[(size_t)c * L + ll] * sc;
    }
    float a = 1.f / (1.f + expf(-acc));
    float h = Y[(size_t)c * L + l] * sc;
    O[(size_t)c * L + l] = fmaxf(h * a, 0.f);
}

// BatchNorm stats (optionally over relu(x)) and in-place apply
__global__ void bn_stats(const float* __restrict__ X, float* __restrict__ sum,
                         float* __restrict__ sumsq, int L, int preRelu)
{
    __shared__ float rs[256], rq[256];
    const float* row = X + (size_t)blockIdx.x * L;
    float s = 0.f, q = 0.f;
    for (int l = threadIdx.x; l < L; l += 256) {
        float v = row[l];
        if (preRelu) v = fmaxf(v, 0.f);
        s += v; q += v * v;
    }
    rs[threadIdx.x] = s; rq[threadIdx.x] = q; __syncthreads();
    for (int k = 128; k > 0; k >>= 1) {
        if (threadIdx.x < k) { rs[threadIdx.x] += rs[threadIdx.x + k];
                               rq[threadIdx.x] += rq[threadIdx.x + k]; }
        __syncthreads();
    }
    if (threadIdx.x == 0) { sum[blockIdx.x] = rs[0]; sumsq[blockIdx.x] = rq[0]; }
}

__global__ void bn_apply(float* __restrict__ X, const float* __restrict__ sum,
                         const float* __restrict__ sumsq,
                         const float* __restrict__ g, const float* __restrict__ b,
                         int C, int L, int preRelu)
{
    long i = (long)blockIdx.x * blockDim.x + threadIdx.x;
    if (i >= (long)C * L) return;
    int c = (int)(i / L);
    float v = X[i];
    if (preRelu) v = fmaxf(v, 0.f);
    float m  = sum[c] / (float)L;
    float vr = sumsq[c] / (float)L - m * m;
    X[i] = (v - m) * rsqrtf(vr + 1e-5f) * g[c] + b[c];
}

// RPN cls/bbox heads, directly in anchor-major (l*5+a) layout
__global__ void rpn_heads(const float* __restrict__ R,
                          const float* __restrict__ cw, const float* __restrict__ cb,
                          const float* __restrict__ bw, const float* __restrict__ bb,
                          float* __restrict__ SC, float* __restrict__ D0,
                          float* __restrict__ D1, int L)
{
    int i = blockIdx.x * blockDim.x + threadIdx.x;
    if (i >= NANCH * L) return;
    int a = i % NANCH, l = i / NANCH;
    float accC = cb[a], acc0 = bb[2 * a], acc1 = bb[2 * a + 1];
#pragma unroll
    for (int c = 0; c < 16; ++c) {
        float r = R[(size_t)c * L + l];
        accC += cw[a * 16 + c] * r;
        acc0 += bw[(2 * a) * 16 + c] * r;
        acc1 += bw[(2 * a + 1) * 16 + c] * r;
    }
    SC[i] = 1.f / (1.f + expf(-accC));
    D0[i] = acc0;
    D1[i] = acc1;
}

// ---------------------------------------------------------------------------
// top-k (6000) radix select over sortable-uint keys, then compact + bitonic sort
// ---------------------------------------------------------------------------
__device__ inline unsigned f2key(float f) {
    unsigned u = __float_as_uint(f);
    return (u & 0x80000000u) ? ~u : (u | 0x80000000u);
}
__device__ inline float key2f(unsigned k) {
    return __uint_as_float((k & 0x80000000u) ? (k & 0x7FFFFFFFu) : ~k);
}

__global__ void rs_init(unsigned* state, unsigned* hist, unsigned* cntAB,
                        unsigned* candKey, unsigned* candIdx)
{
    int t = blockIdx.x * blockDim.x + threadIdx.x;
    if (t < 256) hist[t] = 0u;
    if (t == 0) { state[0] = 0u; state[1] = PRE_N; state[2] = 0u;
                  cntAB[0] = 0u; cntAB[1] = 0u; }
    if (t >= PRE_N && t < SORT_N) { candKey[t] = 0u; candIdx[t] = 0u; }
}

__global__ void rs_hist(const float* __restrict__ sc, int N, int shift,
                        const unsigned* __restrict__ state, unsigned* __restrict__ hist)
{
    unsigned prefix = state[0];
    int stride = gridDim.x * blockDim.x;
    for (int i = blockIdx.x * blockDim.x + threadIdx.x; i < N; i += stride) {
        unsigned key = f2key(sc[i]);
        bool match = (shift == 24) ? true : ((key >> (shift + 8)) == prefix);
        if (match) atomicAdd(&hist[(key >> shift) & 255u], 1u);
    }
}

__global__ void rs_scan(unsigned* state, unsigned* hist)
{
    unsigned remK = state[1], acc = 0u;
    int d;
    for (d = 255; d > 0; --d) {
        unsigned c = hist[d];
        if (acc + c >= remK) break;
        acc += c;
    }
    state[0] = (state[0] << 8) | (unsigned)d;
    state[1] = remK - acc;
    state[2] += acc;
    for (int b = 0; b < 256; ++b) hist[b] = 0u;
}

__global__ void rs_compact(const float* __restrict__ sc, int N,
                           const unsigned* __restrict__ state, unsigned* cntAB,
                           unsigned* __restrict__ candKey, unsigned* __restrict__ candIdx)
{
    unsigned T = state[0], G = state[2];
    int stride = gridDim.x * blockDim.x;
    for (int i = blockIdx.x * blockDim.x + threadIdx.x; i < N; i += stride) {
        unsigned key = f2key(sc[i]);
        if (key > T) {
            unsigned p = atomicAdd(&cntAB[0], 1u);
            if (p < PRE_N) { candKey[p] = key; candIdx[p] = (unsigned)i; }
        } else if (key == T) {
            unsigned q = atomicAdd(&cntAB[1], 1u);
            unsigned p = G + q;
            if (p < PRE_N) { candKey[p] = key; candIdx[p] = (unsigned)i; }
        }
    }
}

__global__ void bitonic_stage(unsigned* __restrict__ key, unsigned* __restrict__ idx,
                              int k, int j)
{
    int t = blockIdx.x * blockDim.x + threadIdx.x;       // SORT_N/2 threads
    int i = (t / j) * (2 * j) + (t % j);
    int p = i + j;
    bool desc = ((i & k) == 0);
    unsigned ki = key[i], kp = key[p];
    bool sw = desc ? (ki < kp) : (ki > kp);
    if (sw) {
        key[i] = kp; key[p] = ki;
        unsigned tmp = idx[i]; idx[i] = idx[p]; idx[p] = tmp;
    }
}

// Anchor decode of the sorted 6000 proposals
__global__ void decode6k(const unsigned* __restrict__ candKey,
                         const unsigned* __restrict__ candIdx,
                         const float* __restrict__ D0, const float* __restrict__ D1,
                         float* __restrict__ scS, float* __restrict__ bsA,
                         float* __restrict__ beA, int L)
{
    int p = blockIdx.x * blockDim.x + threadIdx.x;
    if (p >= PRE_N) return;
    const float widths[NANCH] = {4.f, 8.f, 16.f, 32.f, 64.f};
    unsigned idx = candIdx[p];
    int a = (int)(idx % NANCH), l = (int)(idx / NANCH);
    float aw = widths[a];
    float dx = D0[idx];
    float dw = fminf(10.f, fmaxf(-10.f, D1[idx]));
    float pc = dx * aw + (float)l;
    float pw = aw * expf(dw);
    float mx = (float)(L - 1);
    bsA[p] = fminf(mx, fmaxf(0.f, pc - 0.5f * (pw - 1.f)));
    beA[p] = fminf(mx, fmaxf(0.f, pc + 0.5f * (pw - 1.f)));
    scS[p] = key2f(candKey[p]);
}

// NMS suppression-bit matrix: mask[i][w] bit b set iff j=w*32+b > i and IoU > 0.7
__global__ void nms_mask(const float* __restrict__ bs, const float* __restrict__ be,
                         unsigned* __restrict__ mask)
{
    int i = blockIdx.x, w = threadIdx.x;
    float bsi = bs[i], bei = be[i];
    float li = bei - bsi + 1.f;
    unsigned bits = 0u;
    int j0 = w * 32;
#pragma unroll 4
    for (int b = 0; b < 32; ++b) {
        int j = j0 + b;
        if (j < PRE_N && j > i) {
            float inter = fmaxf(0.f, fminf(bei, be[j]) - fmaxf(bsi, bs[j]) + 1.f);
            float lj = be[j] - bs[j] + 1.f;
            float iou = inter / (li + lj - inter);
            if (iou > 0.7f) bits |= 1u << b;
        }
    }
    mask[(size_t)i * NMS_W + w] = bits;
}

// Sequential NMS bit-sweep (the unavoidable serial dependency chain)
__global__ void nms_scan(const unsigned* __restrict__ mask, unsigned* __restrict__ keepOut)
{
    __shared__ unsigned keep[NMS_W];
    int t = threadIdx.x;
    unsigned init = 0u;
    for (int b = 0; b < 32; ++b) { int j = t * 32 + b; if (j < PRE_N) init |= 1u << b; }
    keep[t] = init;
    __syncthreads();
    for (int i = 0; i < PRE_N; ++i) {
        unsigned w = keep[i >> 5];
        bool alive = (w >> (i & 31)) & 1u;
        __syncthreads();
        if (alive) keep[t] &= ~mask[(size_t)i * NMS_W + t];
        __syncthreads();
    }
    keepOut[t] = keep[t];
}

// Compact first 300 kept (score order) into [score,start,end] rows; zero rest
__global__ void nms_gather(const unsigned* __restrict__ keep,
                           const float* __restrict__ sc, const float* __restrict__ bs,
                           const float* __restrict__ be, float* __restrict__ out)
{
    if (threadIdx.x != 0 || blockIdx.x != 0) return;
    for (int i = 0; i < POST_N * 3; ++i) out[i] = 0.f;
    int rank = 0;
    for (int i = 0; i < PRE_N && rank < POST_N; ++i) {
        if ((keep[i >> 5] >> (i & 31)) & 1u) {
            out[rank * 3 + 0] = sc[i];
            out[rank * 3 + 1] = bs[i];
            out[rank * 3 + 2] = be[i];
            ++rank;
        }
    }
}

// ---------------------------------------------------------------------------
// Host orchestration
// ---------------------------------------------------------------------------
struct AdsP {
    const float *at_b, *at_w, *bn_b, *bn_g, *dw_b, *dw_w,
                *pw_b, *pw_w, *se_b1, *se_b2, *se_w1, *se_w2;
};

static AdsP grab(void* const* d_in, int base) {
    AdsP p;
    const float** f = (const float**)&p;
    for (int i = 0; i < 12; ++i) f[i] = (const float*)d_in[base + i];
    return p;
}

template<int CIN, int COUT>
static void run_ads(const float* x0, int c0, const float* x1, int c1,
                    int dil, const AdsP& p,
                    float* Tb, float* Yb, float* Ob,
                    float* mean, float* sse, float* bsum, float* bsq,
                    int L, hipStream_t stream)
{
    long tot  = (long)CIN  * L;
    long tot2 = (long)COUT * L;
    dw3_conv<<<dim3((unsigned)((tot + 255) / 256)), 256, 0, stream>>>(
        x0, c0, x1, c1, p.dw_w, p.dw_b, Tb, dil, L, 1);
    pw_conv_wmma<CIN, COUT><<<dim3(L / 64, (COUT + 15) / 16), 128, 0, stream>>>(
        Tb, p.pw_w, p.pw_b, Yb, L);
    chan_mean<<<COUT, 256, 0, stream>>>(Yb, mean, L);
    se_mlp<<<1, 64, 0, stream>>>(mean, p.se_w1, p.se_b1, p.se_w2, p.se_b2, sse, COUT, 2);
    attn_apply<<<dim3((unsigned)((tot2 + 255) / 256)), 256, 0, stream>>>(
        Yb, sse, p.at_w, p.at_b, Ob, COUT, L);
    bn_stats<<<COUT, 256, 0, stream>>>(Ob, bsum, bsq, L, 0);
    bn_apply<<<dim3((unsigned)((tot2 + 255) / 256)), 256, 0, stream>>>(
        Ob, bsum, bsq, p.bn_g, p.bn_b, COUT, L, 0);
}

extern "C" void kernel_launch(void* const* d_in, const int* in_sizes, int n_in,
                              void* d_out, int out_size, void* d_ws, size_t ws_size,
                              hipStream_t stream)
{
    (void)in_sizes; (void)n_in; (void)out_size; (void)ws_size;
    const int L = SEQ_L;
    const int N = NANCH * L;

    // ---- inputs (alphabetical tree-flatten of the params dict, then sequence)
    const float* bbox_b = (const float*)d_in[0];
    const float* bbox_w = (const float*)d_in[1];
    const float* cls_b  = (const float*)d_in[2];
    const float* cls_w  = (const float*)d_in[3];
    AdsP dec[5], enc[5];
    for (int i = 0; i < 5; ++i) dec[i] = grab(d_in, 4 + 12 * i);
    for (int i = 0; i < 5; ++i) enc[i] = grab(d_in, 64 + 12 * i);
    const float* rpn_bn_b = (const float*)d_in[124];
    const float* rpn_bn_g = (const float*)d_in[125];
    const float* rpn_dw_b = (const float*)d_in[126];
    const float* rpn_dw_w = (const float*)d_in[127];
    const float* rpn_pw_b = (const float*)d_in[128];
    const float* rpn_pw_w = (const float*)d_in[129];
    const float* seq      = (const float*)d_in[130];

    // ---- workspace layout (floats)
    float* W  = (float*)d_ws;
    float* A  = W;                      // 64*L ping
    float* Bb = W + (size_t)64  * L;    // 64*L pong
    float* S0 = W + (size_t)128 * L;    // 32*L
    float* S1 = W + (size_t)160 * L;    // 16*L
    float* S2 = W + (size_t)176 * L;    //  8*L
    float* S3 = W + (size_t)184 * L;    //  4*L
    float* SM = W + (size_t)188 * L;
    float* mean = SM;        float* sse  = SM + 64;
    float* bsum = SM + 128;  float* bsq  = SM + 192;
    float* scS  = SM + 256;
    float* bsA  = scS + 6144;
    float* beA  = bsA + 6144;
    unsigned* U = (unsigned*)(beA + 6144);
    unsigned* hist    = U;
    unsigned* state   = U + 256;
    unsigned* cntAB   = U + 264;
    unsigned* candKey = U + 272;
    unsigned* candIdx = candKey + SORT_N;
    unsigned* keepW   = candIdx + SORT_N;
    unsigned* mask    = keepW + NMS_W;    // 6000*192 words
    float* SC = A;                 // heads outputs reuse ping buffer (dead region)
    float* D0 = A + (size_t)5  * L;
    float* D1 = A + (size_t)10 * L;

    // ---- encoder (ping-pong A/B; skips copied aside)
    run_ads<14, 32>(seq, 14, nullptr, 0, 1, enc[0], A, Bb, A,  mean, sse, bsum, bsq, L, stream);
    hipMemcpyAsync(S0, A,  (size_t)32 * L * 4, hipMemcpyDeviceToDevice, stream);
    run_ads<32, 16>(A, 32, nullptr, 0, 1, enc[1], Bb, A, Bb,  mean, sse, bsum, bsq, L, stream);
    hipMemcpyAsync(S1, Bb, (size_t)16 * L * 4, hipMemcpyDeviceToDevice, stream);
    run_ads<16, 8>(Bb, 16, nullptr, 0, 2, enc[2], A, Bb, A,   mean, sse, bsum, bsq, L, stream);
    hipMemcpyAsync(S2, A,  (size_t)8  * L * 4, hipMemcpyDeviceToDevice, stream);
    run_ads<8, 4>(A, 8, nullptr, 0, 2, enc[3], Bb, A, Bb,     mean, sse, bsum, bsq, L, stream);
    hipMemcpyAsync(S3, Bb, (size_t)4  * L * 4, hipMemcpyDeviceToDevice, stream);
    run_ads<4, 2>(Bb, 4, nullptr, 0, 3, enc[4], A, Bb, A,     mean, sse, bsum, bsq, L, stream);

    // ---- decoder with U-Net skip concats (dual-source depthwise)
    run_ads<2, 4>(A, 2, nullptr, 0, 3, dec[0], Bb, A, Bb,     mean, sse, bsum, bsq, L, stream);
    run_ads<8, 8>(Bb, 4, S3, 4, 2,  dec[1], A, Bb, A,         mean, sse, bsum, bsq, L, stream);
    run_ads<16, 16>(A, 8, S2, 8, 2,  dec[2], Bb, A, Bb,       mean, sse, bsum, bsq, L, stream);
    run_ads<32, 32>(Bb, 16, S1, 16, 1, dec[3], A, Bb, A,      mean, sse, bsum, bsq, L, stream);
    run_ads<64, 32>(A, 32, S0, 32, 1, dec[4], Bb, A, Bb,      mean, sse, bsum, bsq, L, stream);

    // ---- RPN head: dw (no relu) -> pw (WMMA) -> relu+BN -> cls/bbox heads
    long t32 = (long)32 * L;
    dw3_conv<<<dim3((unsigned)((t32 + 255) / 256)), 256, 0, stream>>>(
        Bb, 32, nullptr, 0, rpn_dw_w, rpn_dw_b, A, 1, L, 0);
    pw_conv_wmma<32, 16><<<dim3(L / 64, 1), 128, 0, stream>>>(A, rpn_pw_w, rpn_pw_b, Bb, L);
    bn_stats<<<16, 256, 0, stream>>>(Bb, bsum, bsq, L, 1);
    long t16 = (long)16 * L;
    bn_apply<<<dim3((unsigned)((t16 + 255) / 256)), 256, 0, stream>>>(
        Bb, bsum, bsq, rpn_bn_g, rpn_bn_b, 16, L, 1);
    rpn_heads<<<(N + 255) / 256, 256, 0, stream>>>(
        Bb, cls_w, cls_b, bbox_w, bbox_b, SC, D0, D1, L);

    // ---- top-6000 radix select (4 x 8-bit digits, device-side state)
    rs_init<<<32, 256, 0, stream>>>(state, hist, cntAB, candKey, candIdx);
    for (int shift = 24; shift >= 0; shift -= 8) {
        rs_hist<<<1024, 256, 0, stream>>>(SC, N, shift, state, hist);
        rs_scan<<<1, 1, 0, stream>>>(state, hist);
    }
    rs_compact<<<1024, 256, 0, stream>>>(SC, N, state, cntAB, candKey, candIdx);

    // ---- sort 8192 (keys descending) via global bitonic stages
    for (int k = 2; k <= SORT_N; k <<= 1)
        for (int j = k >> 1; j > 0; j >>= 1)
            bitonic_stage<<<SORT_N / 2 / 256, 256, 0, stream>>>(candKey, candIdx, k, j);

    // ---- decode, NMS, gather
    decode6k<<<(PRE_N + 255) / 256, 256, 0, stream>>>(candKey, candIdx, D0, D1,
                                                      scS, bsA, beA, L);
    nms_mask<<<PRE_N, NMS_W, 0, stream>>>(bsA, beA, mask);
    nms_scan<<<1, NMS_W, 0, stream>>>(mask, keepW);
    nms_gather<<<1, 1, 0, stream>>>(keepW, scS, bsA, beA, (float*)d_out);
}